// TransferFunctionDictionary_75909251990074
// MI455X (gfx1250) — compile-verified
//
#include <hip/hip_runtime.h>
#include <hip/hip_bf16.h>

typedef float v2f __attribute__((ext_vector_type(2)));
typedef float v8f __attribute__((ext_vector_type(8)));

#ifndef M_PI_F
#define M_PI_F 3.14159265358979323846f
#endif

#define NFREQ 16
#define MAX_FRAMES 256

// One block = (one atom, one sample-chunk). 256 threads = 8 waves.
// Each wave processes 16-sample tiles; per tile it builds the 16(partials) x 16(samples)
// matrix of sin(freq*(t+1))*amp^2*dec(t) values directly in WMMA B-fragment layout and
// reduces over partials with 4 chained v_wmma_f32_16x16x4_f32 (A == 1/16 everywhere).
__global__ __launch_bounds__(256) void tfd_synth_kernel(
    const float* __restrict__ osc,
    const float* __restrict__ amp,
    const float* __restrict__ decay,
    const int* __restrict__ p_nsamples,
    const int* __restrict__ p_nframes,
    float* __restrict__ out)
{
    const int atom   = blockIdx.x;
    const int chunk  = blockIdx.y;
    const int nchunk = gridDim.y;

    const int n_samples = p_nsamples[0];
    int n_frames = p_nframes[0];
    if (n_frames > MAX_FRAMES) n_frames = MAX_FRAMES;

    __shared__ float ldsFreq[NFREQ];
    __shared__ float ldsAmp2[NFREQ];
    __shared__ float ldsDec[MAX_FRAMES * NFREQ];   // ldsDec[f*16 + fr] = d^(f+1)

    const int tid = threadIdx.x;

    // Per-partial parameters: freq = sigmoid(osc)^2 * pi, amp2 = amp^2
    if (tid < NFREQ) {
        float o = osc[atom * NFREQ + tid];
        float s = 1.0f / (1.0f + __expf(-o));
        ldsFreq[tid] = s * s * M_PI_F;
        float a = amp[atom * NFREQ + tid];
        ldsAmp2[tid] = a * a;
    }
    __syncthreads();

    // Per-frame decay table: d = 0.7 + sigmoid(decay)*0.299999 ; frame f -> d^(f+1)
    for (int idx = tid; idx < n_frames * NFREQ; idx += blockDim.x) {
        int f  = idx >> 4;
        int fr = idx & 15;
        float dc = decay[atom * NFREQ + fr];
        float sg = 1.0f / (1.0f + __expf(-dc));
        float d  = 0.7f + sg * 0.299999f;
        ldsDec[idx] = __expf((float)(f + 1) * __logf(d));
    }
    __syncthreads();

    const int lane  = tid & 31;
    const int wave  = tid >> 5;
    const int nwv   = blockDim.x >> 5;        // 8 waves
    const int half  = lane >> 4;              // which 16-lane half
    const int tcol  = lane & 15;              // B/D column (sample within tile)

    // Preload the 8 partials this lane touches: fr = 4c + j + 2*half, c in 0..3, j in 0..1
    float fq[8], am[8];
#pragma unroll
    for (int c = 0; c < 4; ++c)
#pragma unroll
        for (int j = 0; j < 2; ++j) {
            int fr = 4 * c + j + 2 * half;
            fq[c * 2 + j] = ldsFreq[fr];
            am[c * 2 + j] = ldsAmp2[fr];
        }

    // A-matrix (16x4 f32): every element = 1/16 -> D row = mean over partials
    v2f afrag;
    afrag.x = 0.0625f;
    afrag.y = 0.0625f;

    const float scale  = (float)n_frames / (float)n_samples;  // T / n_out
    const int   ntiles = n_samples >> 4;
    const int   worker = chunk * nwv + wave;
    const int   nworkers = nchunk * nwv;
    float* __restrict__ outA = out + (size_t)atom * (size_t)n_samples;

    for (int tile = worker; tile < ntiles; tile += nworkers) {
        const int   t    = (tile << 4) + tcol;
        // linear-interp position into the frame table (align_corners=False)
        float pos = ((float)t + 0.5f) * scale - 0.5f;
        pos = fminf(fmaxf(pos, 0.0f), (float)(n_frames - 1));
        const int   i0 = (int)pos;                       // pos >= 0 -> trunc == floor
        const int   i1 = min(i0 + 1, n_frames - 1);
        const float w  = pos - (float)i0;
        const float tp1 = (float)(t + 1);

        v8f acc = {};  // C starts at zero

#pragma unroll
        for (int c = 0; c < 4; ++c) {
            v2f b;
#pragma unroll
            for (int j = 0; j < 2; ++j) {
                const int k  = c * 2 + j;
                const int fr = 4 * c + j + 2 * half;
                float d0 = ldsDec[(i0 << 4) + fr];
                float d1 = ldsDec[(i1 << 4) + fr];
                float dec = d0 + w * (d1 - d0);
                float sg  = __sinf(fq[k] * tp1);
                float val = sg * am[k] * dec;
                if (j == 0) b.x = val; else b.y = val;
            }
            // D = A(1/16) x B + C   -- chained accumulation over the 4 K-chunks
            acc = __builtin_amdgcn_wmma_f32_16x16x4_f32(
                /*neg_a=*/false, afrag, /*neg_b=*/false, b,
                /*c_mod=*/(short)0, acc, /*reuse_a=*/false, /*reuse_b=*/false);
        }

        // D VGPR0, lanes 0..15 = row M=0, N=lane -> the 16 reduced outputs
        if (half == 0) {
            outA[(tile << 4) + tcol] = acc[0];
        }
    }
}

extern "C" void kernel_launch(void* const* d_in, const int* in_sizes, int n_in,
                              void* d_out, int out_size, void* d_ws, size_t ws_size,
                              hipStream_t stream) {
    (void)n_in; (void)d_ws; (void)ws_size; (void)out_size;
    const float* osc   = (const float*)d_in[0];
    const float* amp   = (const float*)d_in[1];
    const float* decay = (const float*)d_in[2];
    const int*   nsmp  = (const int*)d_in[3];
    const int*   nfrm  = (const int*)d_in[4];

    const int A = in_sizes[0] / NFREQ;   // 128 atoms for the reference setup
    dim3 grid(A, 16, 1);                 // ~2048 blocks to fill the chip
    dim3 block(256, 1, 1);               // 8 wave32s
    tfd_synth_kernel<<<grid, block, 0, stream>>>(osc, amp, decay, nsmp, nfrm,
                                                 (float*)d_out);
}